// FSQuantizer_5162550690363
// MI455X (gfx1250) — compile-verified
//
#include <hip/hip_runtime.h>
#include <hip/hip_bf16.h>

typedef __attribute__((ext_vector_type(2))) float v2f;
typedef __attribute__((ext_vector_type(8))) float v8f;

#define BB 32
#define CC 512
#define TT 4096
#define NN (BB * TT)        // 131072 tokens
#define EE 4
#define NE_BINS 1000
#define BCT (BB * CC * TT)  // 67108864

// ---------------------------------------------------------------------------
// Pass 0: zero the histogram bins (harness does not re-poison between replays)
// ---------------------------------------------------------------------------
__global__ void k_zero_counts(unsigned* __restrict__ counts) {
    int i = blockIdx.x * blockDim.x + threadIdx.x;
    if (i < NE_BINS) counts[i] = 0u;
}

// ---------------------------------------------------------------------------
// Pass 1: h = x @ W_in + b_in  (x[n,c] = z[b,c,t], n = b*T + t)
// Each thread handles 4 consecutive t (float4 coalesced loads of z).
// Also produces per-block partial sums of h and h^2 for BatchNorm stats.
// grid = 128 blocks x 256 threads  (128*256*4 = 131072 tokens)
// ---------------------------------------------------------------------------
__global__ __launch_bounds__(256) void k_proj_in(
    const float* __restrict__ z, const float* __restrict__ W_in,
    const float* __restrict__ b_in, float* __restrict__ h,
    float* __restrict__ partA) {
    const int tid = threadIdx.x;
    const int n0  = (blockIdx.x * 256 + tid) * 4;  // first of 4 tokens
    const int b   = n0 >> 12;                      // / TT
    const int t0  = n0 & (TT - 1);
    const float* zb = z + (size_t)b * CC * TT + t0;

    const float bi0 = b_in[0], bi1 = b_in[1], bi2 = b_in[2], bi3 = b_in[3];
    float acc[4][4];
#pragma unroll
    for (int k = 0; k < 4; ++k) {
        acc[k][0] = bi0; acc[k][1] = bi1; acc[k][2] = bi2; acc[k][3] = bi3;
    }

    const float4* W4 = (const float4*)W_in;  // W_in is [512,4] row-major
#pragma unroll 4
    for (int c = 0; c < CC; ++c) {
        const float4 w  = W4[c];                              // uniform -> SGPR
        const float4 zv = *(const float4*)(zb + (size_t)c * TT);  // 4 tokens
        acc[0][0] += zv.x * w.x; acc[0][1] += zv.x * w.y;
        acc[0][2] += zv.x * w.z; acc[0][3] += zv.x * w.w;
        acc[1][0] += zv.y * w.x; acc[1][1] += zv.y * w.y;
        acc[1][2] += zv.y * w.z; acc[1][3] += zv.y * w.w;
        acc[2][0] += zv.z * w.x; acc[2][1] += zv.z * w.y;
        acc[2][2] += zv.z * w.z; acc[2][3] += zv.z * w.w;
        acc[3][0] += zv.w * w.x; acc[3][1] += zv.w * w.y;
        acc[3][2] += zv.w * w.z; acc[3][3] += zv.w * w.w;
    }

    float s[8] = {0, 0, 0, 0, 0, 0, 0, 0};  // [0..3] sum, [4..7] sumsq
#pragma unroll
    for (int k = 0; k < 4; ++k) {
        float4 hv = {acc[k][0], acc[k][1], acc[k][2], acc[k][3]};
        ((float4*)h)[n0 + k] = hv;
#pragma unroll
        for (int e = 0; e < 4; ++e) {
            s[e]     += acc[k][e];
            s[4 + e] += acc[k][e] * acc[k][e];
        }
    }

    __shared__ float red[256];
#pragma unroll
    for (int j = 0; j < 8; ++j) {
        red[tid] = s[j];
        __syncthreads();
        for (int off = 128; off; off >>= 1) {
            if (tid < off) red[tid] += red[tid + off];
            __syncthreads();
        }
        if (tid == 0) partA[blockIdx.x * 8 + j] = red[0];
        __syncthreads();
    }
}

// ---------------------------------------------------------------------------
// Pass 2: finalize BN stats -> fused per-channel scale/bias (8 floats)
// scale = rsqrt(var+eps)*gamma ; bias = beta - mu*scale
// ---------------------------------------------------------------------------
__global__ void k_stats(const float* __restrict__ partA,
                        const float* __restrict__ gamma,
                        const float* __restrict__ beta,
                        float* __restrict__ sb, int nblk) {
    const int j = threadIdx.x;
    __shared__ float S[8];
    if (j < 8) {
        float s = 0.f;
        for (int i = 0; i < nblk; ++i) s += partA[i * 8 + j];
        S[j] = s;
    }
    __syncthreads();
    if (j < 4) {
        const float mu  = S[j] / (float)NN;
        const float var = S[4 + j] / (float)NN - mu * mu;  // biased, like jnp.var
        const float sc  = rsqrtf(var + 1e-5f) * gamma[j];
        sb[j]     = sc;
        sb[4 + j] = beta[j] - mu * sc;
    }
}

// ---------------------------------------------------------------------------
// Pass 3: BN apply + FSQ bound/round -> codes zq [N,4], histogram of indices.
// LEVELS = [8,5,5,5]; half_w = [4,2,2,2]; basis = [1,8,40,200]
// half_l = (L-1)*(1-1e-3)/2 ; offset = 0.5 for even L ; shift = tan(offset/half_l)
// ---------------------------------------------------------------------------
__global__ __launch_bounds__(256) void k_quant(
    const float* __restrict__ h, const float* __restrict__ sb,
    float* __restrict__ zq, unsigned* __restrict__ counts) {
    const int n = blockIdx.x * 256 + threadIdx.x;
    const float4 hv = ((const float4*)h)[n];

    const float half_l[4] = {3.4965f, 1.998f, 1.998f, 1.998f};
    const float offs[4]   = {0.5f, 0.f, 0.f, 0.f};
    const float inv_hw[4] = {0.25f, 0.5f, 0.5f, 0.5f};

    float hh[4];
    hh[0] = hv.x * sb[0] + sb[4];
    hh[1] = hv.y * sb[1] + sb[5];
    hh[2] = hv.z * sb[2] + sb[6];
    hh[3] = hv.w * sb[3] + sb[7];

    float q[4];
#pragma unroll
    for (int e = 0; e < 4; ++e) {
        const float shift   = tanf(offs[e] / half_l[e]);
        const float bounded = tanhf(hh[e] + shift) * half_l[e] - offs[e];
        q[e] = rintf(bounded);  // round-to-nearest-even == jnp.round
    }

    float4 zv = {q[0] * inv_hw[0], q[1] * inv_hw[1], q[2] * inv_hw[2], q[3] * inv_hw[3]};
    ((float4*)zq)[n] = zv;

    const int idx = (int)(q[0] + 4.f) + 8 * (int)(q[1] + 2.f) +
                    40 * (int)(q[2] + 2.f) + 200 * (int)(q[3] + 2.f);
    atomicAdd(&counts[idx], 1u);
}

// ---------------------------------------------------------------------------
// Pass 4: project_out via V_WMMA_F32_16X16X4_F32 (transposed tile), fused
// commit-loss partials.  D(16ch x 16tok) = A(16ch x 4) * B(4 x 16tok).
//   A = W_out^T tile, B = zq^T tile  -> D lanes vary over t: coalesced stores.
// grid = N/16 token-tile blocks x 256 threads (8 waves); each wave covers
// 4 channel tiles of 16 -> 8*4*16 = 512 channels per block.
// ---------------------------------------------------------------------------
__global__ __launch_bounds__(256) void k_proj_out(
    const float* __restrict__ zq, const float* __restrict__ W_out,
    const float* __restrict__ b_out, const float* __restrict__ z,
    float* __restrict__ out, float* __restrict__ partC) {
    const int tid     = threadIdx.x;
    const int lane    = tid & 31;
    const int wid     = tid >> 5;            // 0..7
    const int tokbase = blockIdx.x * 16;
    const int b       = tokbase >> 12;       // / TT
    const int tb      = tokbase & (TT - 1);
    const int m       = lane & 15;           // row (A) / col (B,D) within tile
    const int kp      = (lane >> 4) << 1;    // K pair: 0 or 2
    const int hi8     = (lane >> 4) << 3;    // +8 row offset for D in hi half

    // B operand: zq^T (4 x 16 tokens): lane m holds K = kp, kp+1 for token m
    v2f bv;
    bv.x = zq[(size_t)(tokbase + m) * 4 + kp];
    bv.y = zq[(size_t)(tokbase + m) * 4 + kp + 1];

    const size_t zbase = (size_t)b * CC * TT + tb;
    float acc = 0.f;

#pragma unroll
    for (int it = 0; it < 4; ++it) {
        const int cbase = wid * 64 + it * 16;
        // A operand: W_out^T tile (16 channels x 4): W_out is [4, 512] row-major
        v2f av;
        av.x = W_out[(size_t)kp * CC + cbase + m];
        av.y = W_out[(size_t)(kp + 1) * CC + cbase + m];

        v8f cz = {0.f, 0.f, 0.f, 0.f, 0.f, 0.f, 0.f, 0.f};
        // 8 args: (neg_a, A, neg_b, B, c_mod, C, reuse_a, reuse_b)
        v8f d = __builtin_amdgcn_wmma_f32_16x16x4_f32(
            false, av, false, bv, (short)0, cz, false, false);

#pragma unroll
        for (int v = 0; v < 8; ++v) {
            const int   cl   = cbase + v + hi8;       // channel row of D
            const float o    = d[v] + b_out[cl];
            const size_t a   = zbase + (size_t)cl * TT + m;  // t = tb + m
            const float diff = z[a] - o;
            out[a] = o;
            acc += diff * diff;
        }
    }

    __shared__ float red[256];
    red[tid] = acc;
    __syncthreads();
    for (int off = 128; off; off >>= 1) {
        if (tid < off) red[tid] += red[tid + off];
        __syncthreads();
    }
    if (tid == 0) partC[blockIdx.x] = red[0];
}

// ---------------------------------------------------------------------------
// Pass 5: final scalars -> d_out[BCT] = commit_loss, d_out[BCT+1] = perplexity
// ---------------------------------------------------------------------------
__global__ __launch_bounds__(256) void k_final(
    const float* __restrict__ partC, int nblk,
    const unsigned* __restrict__ counts, float* __restrict__ out_scalars) {
    const int tid = threadIdx.x;
    __shared__ float red[256];

    float s = 0.f;
    for (int i = tid; i < nblk; i += 256) s += partC[i];
    red[tid] = s;
    __syncthreads();
    for (int off = 128; off; off >>= 1) {
        if (tid < off) red[tid] += red[tid + off];
        __syncthreads();
    }
    const float commit = red[0] / (float)BCT;
    __syncthreads();

    float p = 0.f;
    for (int i = tid; i < NE_BINS; i += 256) {
        const float em = (float)counts[i] / (float)NN;
        p += em * logf(em + 1e-10f);
    }
    red[tid] = p;
    __syncthreads();
    for (int off = 128; off; off >>= 1) {
        if (tid < off) red[tid] += red[tid + off];
        __syncthreads();
    }
    if (tid == 0) {
        out_scalars[0] = commit;
        out_scalars[1] = expf(-red[0]);
    }
}

// ---------------------------------------------------------------------------
extern "C" void kernel_launch(void* const* d_in, const int* in_sizes, int n_in,
                              void* d_out, int out_size, void* d_ws, size_t ws_size,
                              hipStream_t stream) {
    const float* z     = (const float*)d_in[0];
    const float* W_in  = (const float*)d_in[1];
    const float* b_in  = (const float*)d_in[2];
    const float* gamma = (const float*)d_in[3];
    const float* beta  = (const float*)d_in[4];
    const float* W_out = (const float*)d_in[5];
    const float* b_out = (const float*)d_in[6];
    float* out = (float*)d_out;

    // Workspace layout (~4.2 MB, L2-resident on a 192 MB L2)
    float* h     = (float*)d_ws;                 // NN*4 floats (2 MB)
    float* zq    = h + (size_t)NN * 4;           // NN*4 floats (2 MB)
    float* partA = zq + (size_t)NN * 4;          // 128*8
    float* partC = partA + 128 * 8;              // NN/16 = 8192
    float* sb    = partC + (NN / 16);            // 8 (scale[4], bias[4])
    unsigned* counts = (unsigned*)(sb + 8);      // 1000 bins

    k_zero_counts<<<(NE_BINS + 255) / 256, 256, 0, stream>>>(counts);
    k_proj_in<<<NN / (256 * 4), 256, 0, stream>>>(z, W_in, b_in, h, partA);
    k_stats<<<1, 32, 0, stream>>>(partA, gamma, beta, sb, NN / (256 * 4));
    k_quant<<<NN / 256, 256, 0, stream>>>(h, sb, zq, counts);
    k_proj_out<<<NN / 16, 256, 0, stream>>>(zq, W_out, b_out, z, out, partC);
    k_final<<<1, 256, 0, stream>>>(partC, NN / 16, counts, out + BCT);
}